// DomainNorm_19361712571128
// MI455X (gfx1250) — compile-verified
//
#include <hip/hip_runtime.h>

typedef __attribute__((ext_vector_type(2))) float v2f;
typedef __attribute__((ext_vector_type(4))) float v4f;
typedef __attribute__((ext_vector_type(8))) float v8f;

// Problem constants (fixed by the harness/reference)
constexpr int B_ = 64;
constexpr int H_ = 96;
constexpr int N_ = 16384;
constexpr int E_ = 16;

// ---------------------------------------------------------------------------
// Kernel 1: per-(b,h) row sum over N.  One block per row, 256 threads,
// each thread reads 16 float4 (b128) loads -> LDS tree reduction.
// ---------------------------------------------------------------------------
__global__ void dn_row_reduce(const float* __restrict__ x, float* __restrict__ sums) {
    const int row = blockIdx.x;                       // row = b*H + h, 0..6143
    const v4f* __restrict__ xv =
        reinterpret_cast<const v4f*>(x) + (long long)row * (N_ / 4);
    float s = 0.0f;
#pragma unroll
    for (int j = 0; j < 16; ++j) {
        v4f v = xv[threadIdx.x + j * 256];
        s += (v.x + v.y) + (v.z + v.w);
    }
    __shared__ float red[256];
    red[threadIdx.x] = s;
    __syncthreads();
    for (int off = 128; off > 0; off >>= 1) {
        if (threadIdx.x < off) red[threadIdx.x] += red[threadIdx.x + off];
        __syncthreads();
    }
    if (threadIdx.x == 0) sums[row] = red[0];
}

// ---------------------------------------------------------------------------
// Kernel 2: gate GEMM [64,96]x[96,16] via V_WMMA_F32_16X16X4_F32 + argmax.
// One block of 128 threads = 4 waves; wave w computes rows [16w, 16w+16).
// A-frag (16x4 f32, 2 VGPR): lane<16 -> M=lane, {K, K+1}; lane>=16 -> {K+2,K+3}
// B-frag (4x16 f32, 2 VGPR): mirrored with N=lane&15.
// C/D (16x16 f32, 8 VGPR):  vgpr v -> M = v + (lane>=16 ? 8:0), N = lane&15.
// ---------------------------------------------------------------------------
__global__ void dn_gate_wmma(const float* __restrict__ sums,
                             const float* __restrict__ gate_w,
                             const float* __restrict__ gate_b,
                             const float* __restrict__ gammas,
                             const float* __restrict__ betas,
                             float* __restrict__ scale,
                             float* __restrict__ shift) {
    __shared__ float sc[B_][E_];

    const int lane = threadIdx.x & 31;
    const int wave = threadIdx.x >> 5;
    const int m0   = wave * 16;            // row tile base (batch index)
    const int MN   = lane & 15;            // M for A, N for B/C
    const int khi  = (lane >> 4) ? 2 : 0;  // upper half-wave holds K+2,K+3
    const float invN = 1.0f / (float)N_;

    v8f c = {};
    for (int k0 = 0; k0 < H_; k0 += 4) {
        const int k = k0 + khi;
        v2f a, b;
        a.x = sums[(m0 + MN) * H_ + k]     * invN;   // mean[b][h=k]
        a.y = sums[(m0 + MN) * H_ + k + 1] * invN;   // mean[b][h=k+1]
        b.x = gate_w[k * E_ + MN];                   // W[h=k][e=N]
        b.y = gate_w[(k + 1) * E_ + MN];             // W[h=k+1][e=N]
        c = __builtin_amdgcn_wmma_f32_16x16x4_f32(
                /*neg_a=*/false, a, /*neg_b=*/false, b,
                /*c_mod=*/(short)0, c, /*reuse_a=*/false, /*reuse_b=*/false);
    }

    const int rbase = m0 + ((lane >> 4) ? 8 : 0);
#pragma unroll
    for (int v = 0; v < 8; ++v) sc[rbase + v][MN] = c[v];
    __syncthreads();

    if (threadIdx.x < B_) {
        int   best = 0;
        float bv   = sc[threadIdx.x][0] + gate_b[0];
#pragma unroll
        for (int e = 1; e < E_; ++e) {
            float s = sc[threadIdx.x][e] + gate_b[e];
            if (s > bv) { bv = s; best = e; }   // strict > : first-index tie (jnp.argmax)
        }
        scale[threadIdx.x] = gammas[best];
        shift[threadIdx.x] = betas[best];
    }
}

// ---------------------------------------------------------------------------
// Kernel 3: out = x * scale[b] + shift[b], float4 streaming.
// Blocks iterate in REVERSE address order so the tail of x (still resident in
// the 192MB L2 after kernel 1's forward sweep) is re-read from L2, not HBM.
// Stores are non-temporal (output is never re-read) to avoid evicting x lines.
// 12288 blocks * 256 threads * 8 float4 = 100,663,296 elements exactly.
// ---------------------------------------------------------------------------
__global__ void dn_affine_apply(const float* __restrict__ x,
                                const float* __restrict__ scale,
                                const float* __restrict__ shift,
                                float* __restrict__ out) {
    const int rb = (int)gridDim.x - 1 - (int)blockIdx.x;   // reversed block id
    // 2048 float4 per block; H_*N_/4 = 393216 float4 per batch -> 192 blocks/batch
    const int  bidx  = rb / 192;                           // uniform per block
    const float g    = scale[bidx];
    const float be   = shift[bidx];

    const v4f* __restrict__ xv = reinterpret_cast<const v4f*>(x);
    v4f* __restrict__ ov       = reinterpret_cast<v4f*>(out);

    long long i = (long long)rb * 2048 + threadIdx.x;
#pragma unroll
    for (int j = 0; j < 8; ++j) {
        v4f v = xv[i];
        v.x = fmaf(v.x, g, be);
        v.y = fmaf(v.y, g, be);
        v.z = fmaf(v.z, g, be);
        v.w = fmaf(v.w, g, be);
        __builtin_nontemporal_store(v, &ov[i]);
        i += 256;
    }
}

// ---------------------------------------------------------------------------
extern "C" void kernel_launch(void* const* d_in, const int* in_sizes, int n_in,
                              void* d_out, int out_size, void* d_ws, size_t ws_size,
                              hipStream_t stream) {
    const float* x      = (const float*)d_in[0];
    const float* gammas = (const float*)d_in[1];
    const float* betas  = (const float*)d_in[2];
    const float* gate_w = (const float*)d_in[3];
    const float* gate_b = (const float*)d_in[4];
    float*       out    = (float*)d_out;

    float* sums  = (float*)d_ws;        // B*H = 6144 floats
    float* scale = sums + B_ * H_;      // B floats
    float* shift = scale + B_;          // B floats

    dn_row_reduce<<<B_ * H_, 256, 0, stream>>>(x, sums);
    dn_gate_wmma<<<1, 128, 0, stream>>>(sums, gate_w, gate_b, gammas, betas, scale, shift);
    dn_affine_apply<<<(B_ * H_ * N_) / (256 * 8 * 4), 256, 0, stream>>>(x, scale, shift, out);
}